// MyNLLLoss_82102594830975
// MI455X (gfx1250) — compile-verified
//
#include <hip/hip_runtime.h>
#include <cstdint>
#include <cstddef>

// ---------------------------------------------------------------------------
// MI455X (gfx1250) NLL-of-sequence-prob loss.
// Bound: 1.05 GB streamed once -> ~45us @ 23.3 TB/s. Pure bandwidth kernel.
// CDNA5 paths used: global_load_async_to_lds_b128 sliding-window pipeline
// (ASYNCcnt), s_wait_asynccnt/s_wait_dscnt, wave32 shuffles, and
// V_WMMA_F32_16X16X4_F32 as a full-precision matrix-engine lane reduction.
// ---------------------------------------------------------------------------

#define AS1 __attribute__((address_space(1)))
#define AS3 __attribute__((address_space(3)))

typedef float f32x2 __attribute__((ext_vector_type(2)));
typedef float f32x4 __attribute__((ext_vector_type(4)));
typedef float f32x8 __attribute__((ext_vector_type(8)));
typedef int   v4i   __attribute__((vector_size(16)));   // matches builtin param

#ifndef __has_builtin
#define __has_builtin(x) 0
#endif

#if __has_builtin(__builtin_amdgcn_global_load_async_to_lds_b128)
#define HAVE_ASYNC_LDS 1
#else
#define HAVE_ASYNC_LDS 0
#endif

#if __has_builtin(__builtin_amdgcn_wmma_f32_16x16x4_f32)
#define HAVE_WMMA_F32 1
#else
#define HAVE_WMMA_F32 0
#endif

#define TPB  256   // 8 wave32 per block
#define NBUF 4     // async pipeline depth

// ---------------- wave32 reductions ----------------

__device__ __forceinline__ float wave_max32(float m) {
#pragma unroll
  for (int off = 16; off > 0; off >>= 1)
    m = fmaxf(m, __shfl_xor(m, off, 32));
  return m;
}

// Sum of 32 lane values, fp32. Uses V_WMMA_F32_16X16X4_F32 with B = ones:
// A[m,0] = x[lane m], A[m,2] = x[lane m+16]  =>  D[m,n] = x[m] + x[m+16].
// Each lane then locally sums its 8 D registers (8 distinct M rows) and one
// shfl_xor(16) add yields the full 32-lane sum in every lane.
__device__ __forceinline__ float wave_sum32(float x) {
#if HAVE_WMMA_F32
  f32x2 a; a[0] = x;   a[1] = 0.0f;
  f32x2 b; b[0] = 1.0f; b[1] = 1.0f;
  f32x8 c = {0.f, 0.f, 0.f, 0.f, 0.f, 0.f, 0.f, 0.f};
  f32x8 d = __builtin_amdgcn_wmma_f32_16x16x4_f32(
      /*neg_a=*/false, a, /*neg_b=*/false, b,
      /*c_mod=*/(short)0, c, /*reuse_a=*/false, /*reuse_b=*/false);
  float t = d[0] + d[1] + d[2] + d[3] + d[4] + d[5] + d[6] + d[7];
  t += __shfl_xor(t, 16, 32);
  return t;
#else
#pragma unroll
  for (int off = 16; off > 0; off >>= 1)
    x += __shfl_xor(x, off, 32);
  return x;
#endif
}

// ---------------- online logsumexp ----------------

struct LSE { float m, s; };

__device__ __forceinline__ void lse_add4(LSE& a, f32x4 v) {
  float mv = fmaxf(fmaxf(v[0], v[1]), fmaxf(v[2], v[3]));
  float M  = fmaxf(a.m, mv);
  a.s = a.s * __expf(a.m - M)
      + __expf(v[0] - M) + __expf(v[1] - M)
      + __expf(v[2] - M) + __expf(v[3] - M);
  a.m = M;
}

__device__ __forceinline__ void lse_add1(LSE& a, float v) {
  float M = fmaxf(a.m, v);
  a.s = a.s * __expf(a.m - M) + __expf(v - M);
  a.m = M;
}

__device__ __forceinline__ void lse_merge(LSE& a, float m2, float s2) {
  float M = fmaxf(a.m, m2);
  a.s = a.s * __expf(a.m - M) + s2 * __expf(m2 - M);
  a.m = M;
}

// ---------------- kernel 1: per-row logsumexp + target gather ----------------

__global__ __launch_bounds__(TPB)
void lse_rows_kernel(const float* __restrict__ preds,
                     const long long* __restrict__ targets,
                     float* __restrict__ tok_logp,
                     int V)
{
  const int    row     = blockIdx.x;
  const int    tid     = threadIdx.x;
  const size_t rowbase = (size_t)row * (size_t)V;
  const float* rowp    = preds + rowbase;

  LSE acc; acc.m = -__builtin_inff(); acc.s = 0.0f;

#if HAVE_ASYNC_LDS
  // One block-tile = TPB lanes * 16B = 4KB = 1024 floats. V=32000 -> 31 tiles
  // + 256-float scalar tail. Sliding window of NBUF async tiles in LDS keeps
  // 4 b128-batches in flight per wave with zero VGPR destinations.
  __shared__ f32x4 buf[NBUF * TPB];   // 16 KB
  const int ntiles = V >> 10;         // assumes V >= NBUF*1024 (true: 32000)

  auto issue = [&](int i) {
    const void* g = rowp + ((size_t)i << 10) + ((size_t)tid << 2);
    const void* l = &buf[((i & (NBUF - 1)) << 8) + tid];
    __builtin_amdgcn_global_load_async_to_lds_b128(
        (AS1 v4i*)(unsigned long long)g,
        (AS3 v4i*)(unsigned)(unsigned long long)l,
        0, 0);
  };
  auto consume = [&](int i) {
    f32x4 v = buf[((i & (NBUF - 1)) << 8) + tid];
    lse_add4(acc, v);
  };

  const int npre = (ntiles < NBUF) ? ntiles : NBUF;
  for (int i = 0; i < npre; ++i) issue(i);

  for (int i = 0; i + NBUF < ntiles; ++i) {
    asm volatile("s_wait_asynccnt 3" ::: "memory");   // oldest tile landed
    consume(i);
    asm volatile("s_wait_dscnt 0" ::: "memory");      // LDS read drained
    issue(i + NBUF);                                  // safe to overwrite slot
  }
  { asm volatile("s_wait_asynccnt 3" ::: "memory"); consume(ntiles - 4); }
  { asm volatile("s_wait_asynccnt 2" ::: "memory"); consume(ntiles - 3); }
  { asm volatile("s_wait_asynccnt 1" ::: "memory"); consume(ntiles - 2); }
  { asm volatile("s_wait_asynccnt 0" ::: "memory"); consume(ntiles - 1); }

  const int done = ntiles << 10;
#else
  // Fallback: streaming non-temporal b128 loads straight to VGPRs.
  const f32x4* rowp4 = (const f32x4*)rowp;
  const int    nv4   = V >> 2;
  for (int i = tid; i < nv4; i += TPB) {
    f32x4 v = __builtin_nontemporal_load(rowp4 + i);
    lse_add4(acc, v);
  }
  const int done = nv4 << 2;
#endif

  for (int j = done + tid; j < V; j += TPB)
    lse_add1(acc, rowp[j]);

  // wave reduction: shuffle-max, rescale, WMMA fp32 sum
  float M    = wave_max32(acc.m);
  float Sadj = acc.s * __expf(acc.m - M);
  float S    = wave_sum32(Sadj);

  __shared__ float smM[TPB / 32], smS[TPB / 32];
  const int wave = tid >> 5;
  if ((tid & 31) == 0) { smM[wave] = M; smS[wave] = S; }
  __syncthreads();

  if (tid == 0) {
    LSE r; r.m = -__builtin_inff(); r.s = 0.0f;
#pragma unroll
    for (int w = 0; w < TPB / 32; ++w) lse_merge(r, smM[w], smS[w]);
    const float     lse = r.m + __logf(r.s);
    const long long tgt = targets[row];
    tok_logp[row] = preds[rowbase + (size_t)tgt] - lse;
  }
}

// ---------------- kernel 2: finalize (tiny) ----------------

__global__ __launch_bounds__(128)
void finalize_kernel(const float* __restrict__ tok_logp,
                     const int* __restrict__ ratings,
                     const int* __restrict__ stage,
                     float* __restrict__ out,
                     int S, int L)
{
  const int tid = threadIdx.x;       // tid in [0, B*S)
  const int b   = tid / S;

  const float* t = tok_logp + (size_t)tid * (size_t)L;
  float sum = 0.0f;
  for (int i = 0; i < L; i += 4)
    sum += t[i] + t[i + 1] + t[i + 2] + t[i + 3];

  const float p      = __expf(sum);
  const int   thresh = (stage[0] == 1) ? 4 : 3;
  const float y      = (ratings[b] > thresh) ? 1.0f : 0.0f;
  const float eps    = 1e-10f;
  const float nll    = -(y * __logf(p + eps) + (1.0f - y) * __logf(1.0f - p + eps));

  float wsum = wave_sum32(nll);

  __shared__ float sm[32];
  const int wave = tid >> 5;
  if ((tid & 31) == 0) sm[wave] = wsum;
  __syncthreads();

  if (tid == 0) {
    const int nw = (int)blockDim.x >> 5;
    float total = 0.0f;
    for (int w = 0; w < nw; ++w) total += sm[w];
    // mean_b( sum_s nll / S ) == sum_{b,s} nll / (B*S)
    out[0] = total / (float)blockDim.x;
  }
}

// ---------------- launch ----------------

extern "C" void kernel_launch(void* const* d_in, const int* in_sizes, int n_in,
                              void* d_out, int out_size, void* d_ws, size_t ws_size,
                              hipStream_t stream)
{
  (void)n_in; (void)out_size; (void)ws_size;

  const float*     preds   = (const float*)d_in[0];
  const long long* targets = (const long long*)d_in[1];  // int64
  const int*       ratings = (const int*)d_in[2];        // int32
  const int*       stage   = (const int*)d_in[3];        // scalar

  const int B   = in_sizes[2];              // 8
  const int BSL = in_sizes[1];              // 8192 rows
  const int V   = in_sizes[0] / BSL;        // 32000
  const int L   = 64;                       // per reference shapes
  const int S   = BSL / (B * L);            // 16

  float* tok = (float*)d_ws;                // BSL floats of scratch

  lse_rows_kernel<<<BSL, TPB, 0, stream>>>(preds, targets, tok, V);
  finalize_kernel<<<1, B * S, 0, stream>>>(tok, ratings, stage, (float*)d_out, S, L);
}